// RelPositionMultiHeadAttention_23132693856327
// MI455X (gfx1250) — compile-verified
//
#include <hip/hip_runtime.h>
#include <hip/hip_bf16.h>

// RelPositionMultiHeadAttention for MI455X (gfx1250), fp32 WMMA path.
// L=1024, B=8, E=512, H=8, hd=64.

typedef float v2f __attribute__((ext_vector_type(2)));
typedef float v8f __attribute__((ext_vector_type(8)));

#define L_SEQ 1024
#define BATCH 8
#define EMB   512
#define NH    8
#define HD    64
#define PROWS 2048   // 2L-1 = 2047, padded to 2048 (row 2047 is zero)

// D(16x16,f32) = A(16x4,f32) x B(4x16,f32) + C
// A frag: lane(l<16): row M=l, K={0,1}; lane(l>=16): row M=l-16, K={2,3}
// B frag: lane(l<16): col N=l, K={0,1}; lane(l>=16): col N=l-16, K={2,3}
// C/D:    lane(l<16): N=l, vgpr g -> M=g; lane(l>=16): N=l-16, vgpr g -> M=g+8
__device__ __forceinline__ v8f wmma4(v2f a, v2f b, v8f c) {
  return __builtin_amdgcn_wmma_f32_16x16x4_f32(
      /*neg_a=*/false, a, /*neg_b=*/false, b,
      /*c_mod=*/(short)0, c, /*reuse_a=*/false, /*reuse_b=*/false);
}

// ---------------------------------------------------------------------------
// Kernel 1: qkv = query @ in_proj_w^T + b, scattered to per-head buffers.
// q gets pos_bias_u / pos_bias_v fused (two output copies qu, qv).
// k stored (B,H,L,hd); v stored TRANSPOSED (B,H,hd,L) so that the
// attention O = P@V GEMM gets memory-contiguous K-dim fragments.
// M = L*B = 8192, N = 3E = 1536, K = E = 512.
// grid (512, 24), block 128 (4 waves, one 16x16 tile each).
// ---------------------------------------------------------------------------
__global__ __launch_bounds__(128) void qkv_proj_kernel(
    const float* __restrict__ query,   // (L*B, E) row-major
    const float* __restrict__ w,       // (3E, E)
    const float* __restrict__ bias,    // (3E)
    const float* __restrict__ bu,      // (H*hd)
    const float* __restrict__ bv,      // (H*hd)
    float* __restrict__ qu, float* __restrict__ qv,
    float* __restrict__ kk, float* __restrict__ vt) {
  const int lane = threadIdx.x & 31;
  const int wave = threadIdx.x >> 5;
  const int ln = lane & 15;
  const int hi = lane >> 4;
  const int m0 = blockIdx.x * 16;
  const int n0 = blockIdx.y * 64 + wave * 16;

  v8f acc = {};
  const float* arow = query + (size_t)(m0 + ln) * EMB + 2 * hi;
  const float* brow = w + (size_t)(n0 + ln) * EMB + 2 * hi;
  for (int kc = 0; kc < EMB; kc += 4) {
    v2f a = *(const v2f*)(arow + kc);
    v2f b = *(const v2f*)(brow + kc);
    acc = wmma4(a, b, acc);
  }

  const int col = n0 + ln;        // 0..1535
  const int part = col >> 9;      // 0=q 1=k 2=v
  const int e = col & 511;
  const int h = e >> 6;
  const int d = e & 63;
  const float bval = bias[col];
  const float buv = bu[e];
  const float bvv = bv[e];
#pragma unroll
  for (int g = 0; g < 8; ++g) {
    const int m = m0 + g + 8 * hi;       // row in (L*B)
    const int l = m >> 3;                // B == 8
    const int b_ = m & 7;
    const float val = acc[g] + bval;
    if (part == 0) {
      const size_t idx = (((size_t)(b_ * NH + h)) * L_SEQ + l) * HD + d;
      qu[idx] = val + buv;
      qv[idx] = val + bvv;
    } else if (part == 1) {
      const size_t idx = (((size_t)(b_ * NH + h)) * L_SEQ + l) * HD + d;
      kk[idx] = val;
    } else {
      // transposed: vt[bh][d][l]
      const size_t idx = (((size_t)(b_ * NH + h)) * HD + d) * L_SEQ + l;
      vt[idx] = val;
    }
  }
}

// ---------------------------------------------------------------------------
// Kernel 2: p = pos_emb @ pos_w^T, reshaped to (H, PROWS, hd), pad row zeroed.
// M = 2048 (2047 real), N = 512, K = 512. grid (128, 8), block 128.
// ---------------------------------------------------------------------------
__global__ __launch_bounds__(128) void pos_proj_kernel(
    const float* __restrict__ pe,   // (2047, 512)
    const float* __restrict__ pw,   // (512, 512)
    float* __restrict__ p) {        // (H, 2048, 64)
  const int lane = threadIdx.x & 31;
  const int wave = threadIdx.x >> 5;
  const int ln = lane & 15;
  const int hi = lane >> 4;
  const int m0 = blockIdx.x * 16;
  const int n0 = blockIdx.y * 64 + wave * 16;

  const int mrow = m0 + ln;
  const bool valid = (mrow < 2 * L_SEQ - 1);
  const float msk = valid ? 1.0f : 0.0f;
  const float* arow = pe + (size_t)(valid ? mrow : 0) * EMB + 2 * hi;
  const float* brow = pw + (size_t)(n0 + ln) * EMB + 2 * hi;

  v8f acc = {};
  for (int kc = 0; kc < EMB; kc += 4) {
    v2f a = *(const v2f*)(arow + kc);
    a *= msk;
    v2f b = *(const v2f*)(brow + kc);
    acc = wmma4(a, b, acc);
  }

  const int col = n0 + ln;
  const int h = col >> 6;
  const int d = col & 63;
#pragma unroll
  for (int g = 0; g < 8; ++g) {
    const int m = m0 + g + 8 * hi;
    const float val = (m < 2 * L_SEQ - 1) ? acc[g] : 0.0f;
    p[((size_t)h * PROWS + m) * HD + d] = val;   // m <= 2047 always in-bounds
  }
}

// ---------------------------------------------------------------------------
// Kernel 3: flash-style attention per (b,h, 16-query-row block).
//   score[r][j] = (qu_r . k_j + qv_r . p[j - i + 1023]) * 1/8
//   softmax row-wise in LDS, then O = P @ V with WMMA (V pre-transposed).
// grid (B*H = 64, L/16 = 64), block 128 (4 waves), dynamic LDS ~74 KB.
// ---------------------------------------------------------------------------
__global__ __launch_bounds__(128) void attn_kernel(
    const float* __restrict__ qu, const float* __restrict__ qv,
    const float* __restrict__ kmat, const float* __restrict__ vtmat,
    const float* __restrict__ pmat, float* __restrict__ obuf) {
  extern __shared__ float smem[];
  float* scores = smem;                 // 16*1024 floats (64 KB)
  float* bdst   = smem + 16 * 1024;     // 4 waves * 16*32   (8 KB)
  float* red    = bdst + 4 * 16 * 32;   // 128
  float* rowstat = red + 128;           // 16

  const int tid = threadIdx.x;
  const int lane = tid & 31;
  const int wave = tid >> 5;
  const int ln = lane & 15;
  const int hi = lane >> 4;
  const int bh = blockIdx.x;            // b*8 + h
  const int i0 = blockIdx.y * 16;
  const int h = bh & 7;

  const float* quB = qu + (size_t)bh * (L_SEQ * HD);
  const float* qvB = qv + (size_t)bh * (L_SEQ * HD);
  const float* kB = kmat + (size_t)bh * (L_SEQ * HD);
  const float* vtB = vtmat + (size_t)bh * (HD * L_SEQ);
  const float* pB = pmat + (size_t)h * (PROWS * HD);

  // Resident A fragments for the 16 query rows (K = hd = 64 -> 16 chunks of 4)
  v2f aqu[16], aqv[16];
  {
    const float* qup = quB + (size_t)(i0 + ln) * HD + 2 * hi;
    const float* qvp = qvB + (size_t)(i0 + ln) * HD + 2 * hi;
#pragma unroll
    for (int kk = 0; kk < 16; ++kk) {
      aqu[kk] = *(const v2f*)(qup + 4 * kk);
      aqv[kk] = *(const v2f*)(qvp + 4 * kk);
    }
  }

  float* st = bdst + wave * (16 * 32);

  // ---- Phase 1: scores = (AC + shifted BD) * scale, into LDS ----
  for (int jt = wave; jt < 64; jt += 4) {
    const int j0 = jt * 16;

    v8f ac = {};
    const float* kb = kB + (size_t)(j0 + ln) * HD + 2 * hi;
#pragma unroll
    for (int kk = 0; kk < 16; ++kk)
      ac = wmma4(aqu[kk], *(const v2f*)(kb + 4 * kk), ac);

    // Banded BD: W[r][m] = qv_{i0+r} . p[base + m], m in [0,32)
    // base = j0 - i0 + 1008;  n(r,c) = base + (c - r + 15) in [0,2047]
    const int base = j0 - i0 + (L_SEQ - 16);
    v8f bd0 = {}, bd1 = {};
    const float* pb0 = pB + (size_t)(base + ln) * HD + 2 * hi;
    const float* pb1 = pb0 + 16 * HD;
#pragma unroll
    for (int kk = 0; kk < 16; ++kk) {
      bd0 = wmma4(aqv[kk], *(const v2f*)(pb0 + 4 * kk), bd0);
      bd1 = wmma4(aqv[kk], *(const v2f*)(pb1 + 4 * kk), bd1);
    }

    // Stage wide BD tile in per-wave LDS, then apply the relative shift.
#pragma unroll
    for (int g = 0; g < 8; ++g) {
      const int r = g + 8 * hi;
      st[r * 32 + ln] = bd0[g];
      st[r * 32 + 16 + ln] = bd1[g];
    }
    asm volatile("s_wait_dscnt 0" ::: "memory");  // wave-local LDS RAW
#pragma unroll
    for (int g = 0; g < 8; ++g) {
      const int r = g + 8 * hi;
      const float bdv = st[r * 32 + (ln - r + 15)];
      scores[r * 1024 + j0 + ln] = (ac[g] + bdv) * 0.125f;
    }
  }
  __syncthreads();

  // ---- Phase 2: row-wise softmax (8 threads per row, 128-col chunks) ----
  {
    const int r = tid >> 3;
    const int s = tid & 7;
    float* row = scores + r * 1024 + s * 128;
    float mx = -3.4e38f;
    for (int i = 0; i < 128; ++i) mx = fmaxf(mx, row[i]);
    red[tid] = mx;
    __syncthreads();
    if (s == 0) {
      float m = red[r * 8];
      for (int t = 1; t < 8; ++t) m = fmaxf(m, red[r * 8 + t]);
      rowstat[r] = m;
    }
    __syncthreads();
    const float m = rowstat[r];
    float sum = 0.f;
    for (int i = 0; i < 128; ++i) {
      const float e = __expf(row[i] - m);
      row[i] = e;
      sum += e;
    }
    red[tid] = sum;
    __syncthreads();
    if (s == 0) {
      float t0 = 0.f;
      for (int t = 0; t < 8; ++t) t0 += red[r * 8 + t];
      rowstat[r] = 1.0f / t0;
    }
    __syncthreads();
    const float inv = rowstat[r];
    for (int i = 0; i < 128; ++i) row[i] *= inv;
  }
  __syncthreads();

  // ---- Phase 3: O(16x64) = P(16x1024) @ V(1024x64), one 16x16 tile/wave ----
  // V is pre-transposed (hd, L): B frag B[t][n] = vt[(n0+n)*L + kc + t],
  // so each lane's K-pair is one aligned 8-byte load.
  v8f oacc = {};
  const int n0 = wave * 16;
  const float* arow = scores + ln * 1024 + 2 * hi;
  const float* brow = vtB + (size_t)(n0 + ln) * L_SEQ + 2 * hi;
  for (int kc = 0; kc < 1024; kc += 4) {
    v2f a = *(const v2f*)(arow + kc);
    v2f b = *(const v2f*)(brow + kc);
    oacc = wmma4(a, b, oacc);
  }
  float* ob = obuf + (size_t)bh * (L_SEQ * HD);
#pragma unroll
  for (int g = 0; g < 8; ++g)
    ob[(size_t)(i0 + g + 8 * hi) * HD + n0 + ln] = oacc[g];
}

// ---------------------------------------------------------------------------
// Kernel 4: out = o(L,B,E) @ out_proj_w^T + b.  o read from (B,H,L,hd).
// M = 8192, N = 512, K = 512. grid (512, 8), block 128.
// ---------------------------------------------------------------------------
__global__ __launch_bounds__(128) void out_proj_kernel(
    const float* __restrict__ obuf,   // (B,H,L,hd)
    const float* __restrict__ w,      // (E, E)
    const float* __restrict__ bias,   // (E)
    float* __restrict__ out) {        // (L*B, E)
  const int lane = threadIdx.x & 31;
  const int wave = threadIdx.x >> 5;
  const int ln = lane & 15;
  const int hi = lane >> 4;
  const int m0 = blockIdx.x * 16;
  const int n0 = blockIdx.y * 64 + wave * 16;

  const int m = m0 + ln;
  const int l = m >> 3;
  const int b_ = m & 7;
  const float* base_a = obuf + (size_t)b_ * (NH * L_SEQ * HD) + (size_t)l * HD;
  const float* brow = w + (size_t)(n0 + ln) * EMB + 2 * hi;

  v8f acc = {};
  for (int kc = 0; kc < EMB; kc += 4) {
    const int k = kc + 2 * hi;
    const int hh = k >> 6;
    const int d = k & 63;
    v2f a = *(const v2f*)(base_a + (size_t)hh * (L_SEQ * HD) + d);
    v2f b = *(const v2f*)(brow + kc);
    acc = wmma4(a, b, acc);
  }

  const float bval = bias[n0 + ln];
#pragma unroll
  for (int g = 0; g < 8; ++g)
    out[(size_t)(m0 + g + 8 * hi) * EMB + n0 + ln] = acc[g] + bval;
}

// ---------------------------------------------------------------------------
// Workspace layout (floats):
//   qu   : 4,194,304   (B,H,L,hd)
//   qv   : 4,194,304   (B,H,L,hd)
//   k    : 4,194,304   (B,H,L,hd)
//   vT   : 4,194,304   (B,H,hd,L)
//   p    : 1,048,576   (H, 2048, hd)
//   o    : 4,194,304   (B,H,L,hd)
// total ~84 MB.
// ---------------------------------------------------------------------------
extern "C" void kernel_launch(void* const* d_in, const int* in_sizes, int n_in,
                              void* d_out, int out_size, void* d_ws, size_t ws_size,
                              hipStream_t stream) {
  (void)in_sizes; (void)n_in; (void)out_size; (void)ws_size;
  const float* query   = (const float*)d_in[0];
  const float* pos_emb = (const float*)d_in[1];
  const float* in_w    = (const float*)d_in[2];
  const float* in_b    = (const float*)d_in[3];
  const float* out_w   = (const float*)d_in[4];
  const float* out_b   = (const float*)d_in[5];
  const float* pos_w   = (const float*)d_in[6];
  const float* bu      = (const float*)d_in[7];
  const float* bv      = (const float*)d_in[8];

  float* ws = (float*)d_ws;
  const size_t QSZ = (size_t)BATCH * NH * L_SEQ * HD;   // 4,194,304
  float* qu_ws = ws;
  float* qv_ws = qu_ws + QSZ;
  float* k_ws  = qv_ws + QSZ;
  float* vt_ws = k_ws + QSZ;
  float* p_ws  = vt_ws + QSZ;                           // NH*PROWS*HD
  float* o_ws  = p_ws + (size_t)NH * PROWS * HD;

  const size_t lds_bytes = (16 * 1024 + 4 * 16 * 32 + 128 + 16) * sizeof(float);
  (void)hipFuncSetAttribute(reinterpret_cast<const void*>(attn_kernel),
                            hipFuncAttributeMaxDynamicSharedMemorySize,
                            (int)lds_bytes);

  qkv_proj_kernel<<<dim3(512, 24), 128, 0, stream>>>(
      query, in_w, in_b, bu, bv, qu_ws, qv_ws, k_ws, vt_ws);
  pos_proj_kernel<<<dim3(128, 8), 128, 0, stream>>>(pos_emb, pos_w, p_ws);
  attn_kernel<<<dim3(64, 64), 128, lds_bytes, stream>>>(
      qu_ws, qv_ws, k_ws, vt_ws, p_ws, o_ws);
  out_proj_kernel<<<dim3(512, 8), 128, 0, stream>>>(o_ws, out_w, out_b,
                                                    (float*)d_out);
}